// QHashSoftmax_63136019251231
// MI455X (gfx1250) — compile-verified
//
#include <hip/hip_runtime.h>
#include <math.h>

// Quantized LUT softmax (MASE fixed-point): memory-bound streaming kernel.
// One wave (32 lanes, wave32) per 1024-element row; 8 rows per 256-thread block.
// ~402MB HBM traffic -> ~17us floor at 23.3 TB/s.
//
// LDS LUTs (6KB/block):
//   ldsEf[256] : float exp code E(idx) in [0,127]   (pass-1 gather, exact f32 sum)
//   ldsRE[256] : (E==0) ? 1e9 : exact 1/E           (pass-2 gather)
//   ldsD[1024] : table_div[k] = min(rint(128/k),127)/128
//
// Exact floor(S/E) without integer division:
//   t = S/E has frac part 0 or >= 1/E (E<=127). trunc((S+0.2)*(1/E)) with
//   0.5-ulp 1/E perturbs t by (0.17/E, 0.23/E): positive, < 1/E  => exact floor.

#define ROWS_PER_BLOCK 8

// Q8.4 quantize -> unsigned 8-bit two's-complement index.
// jnp.round is round-half-even -> rintf (v_rndne_f32).
__device__ __forceinline__ int quant_idx(float f) {
    float r = rintf(f * 16.0f);
    r = fmaxf(r, -128.0f);
    r = fminf(r, 127.0f);
    return ((int)r) & 255;
}

__global__ __launch_bounds__(256)
void qsoftmax_lut_kernel(const float* __restrict__ x,
                         const float* __restrict__ scale_p,
                         float* __restrict__ out,
                         int nrows)
{
    __shared__ float ldsEf[256];
    __shared__ float ldsRE[256];
    __shared__ float ldsD[1024];

    // ---- Build LUTs once per block ----
    {
        const float scale = scale_p[0];
        const int t = (int)threadIdx.x;

        // exp-code table (1 expf/thread)
        const int sv = (t >= 128) ? (t - 256) : t;            // two's complement
        float e = expf(((float)sv * 0.0625f) * scale) * 128.0f;
        float E = fminf(rintf(e), 127.0f);                    // exp>0 => no low clamp
        ldsEf[t] = E;
        ldsRE[t] = (E == 0.0f) ? 1e9f : (1.0f / E);           // exact f32 divide

        // reciprocal table: 4 entries/thread; k=0 -> inf -> clamps to 127/128
#pragma unroll
        for (int j = 0; j < 4; ++j) {
            int k = t + j * 256;
            float kr = __builtin_amdgcn_rcpf((float)k);       // 1-ulp rcp is exact
            ldsD[k] = fminf(rintf(128.0f * kr), 127.0f) * 0.0078125f;
        }
    }
    __syncthreads();

    const int wave = (int)threadIdx.x >> 5;
    const int lane = (int)threadIdx.x & 31;
    const int row  = (int)blockIdx.x * ROWS_PER_BLOCK + wave;
    if (row >= nrows) return;

    // 256 float4 per row; lane l touches float4 slots {l, l+32, ..., l+224}:
    // each b128 load/store moves 512B fully-coalesced per wave.
    const float4* __restrict__ rp = (const float4*)x   + (size_t)row * 256 + lane;
    float4* __restrict__       op = (float4*)out       + (size_t)row * 256 + lane;

    __builtin_prefetch((const void*)rp, 0, 0);   // global_prefetch_b8 (gfx1250)

    unsigned packed[8];
    float S = 0.0f;                              // sum of integer-valued floats: exact
#pragma unroll
    for (int it = 0; it < 8; ++it) {
        float4 v = rp[it * 32];
        int i0 = quant_idx(v.x);
        int i1 = quant_idx(v.y);
        int i2 = quant_idx(v.z);
        int i3 = quant_idx(v.w);
        S += (ldsEf[i0] + ldsEf[i1]) + (ldsEf[i2] + ldsEf[i3]);
        packed[it] = (unsigned)i0 | ((unsigned)i1 << 8) |
                     ((unsigned)i2 << 16) | ((unsigned)i3 << 24);
    }

    // wave32 butterfly reduction (f32, exact: S <= 130048 < 2^24)
#pragma unroll
    for (int off = 16; off > 0; off >>= 1)
        S += __shfl_xor(S, off, 32);

    const float Sb = S + 0.2f;                   // floor-division bias (see header)

#pragma unroll
    for (int it = 0; it < 8; ++it) {
        unsigned pk = packed[it];
        float q0 = fminf(Sb * ldsRE[pk & 255u],         1023.0f);
        float q1 = fminf(Sb * ldsRE[(pk >> 8)  & 255u], 1023.0f);
        float q2 = fminf(Sb * ldsRE[(pk >> 16) & 255u], 1023.0f);
        float q3 = fminf(Sb * ldsRE[pk >> 24],          1023.0f);
        float4 o;
        o.x = ldsD[(int)q0];
        o.y = ldsD[(int)q1];
        o.z = ldsD[(int)q2];
        o.w = ldsD[(int)q3];
        op[it * 32] = o;
    }
}

extern "C" void kernel_launch(void* const* d_in, const int* in_sizes, int n_in,
                              void* d_out, int out_size, void* d_ws, size_t ws_size,
                              hipStream_t stream) {
    (void)in_sizes; (void)n_in; (void)d_ws; (void)ws_size;
    const float* x     = (const float*)d_in[0];
    const float* scale = (const float*)d_in[1];
    float* out = (float*)d_out;

    const int nrows  = out_size / 1024;           // last-axis length is 1024
    const int blocks = (nrows + ROWS_PER_BLOCK - 1) / ROWS_PER_BLOCK;
    hipLaunchKernelGGL(qsoftmax_lut_kernel, dim3(blocks), dim3(256), 0, stream,
                       x, scale, out, nrows);
}